// GCGRUCell_19473381720835
// MI455X (gfx1250) — compile-verified
//
#include <hip/hip_runtime.h>
#include <math.h>

// ---------------------------------------------------------------------------
// GCGRU cell, fused, fp32 WMMA (V_WMMA_F32_16X16X4_F32) for MI455X / gfx1250.
// B=4096, d_in=64, U=128, N=36, C=192, c_cat=768.
// Roofline: ~97 GFLOP vs ~190 MB HBM -> memory bound at 23.3 TB/s (~8us floor)
// even with the fp32 WMMA path, so full fp32 precision is kept.
// Data movement uses the CDNA5 async global<->LDS path (ASYNCcnt):
//   - global_load_async_to_lds_b128 to stage adj-powers / x / h into LDS
//   - global_store_async_from_lds_b128 to stream the packed output tile out
// NOTE: async ops address LDS by byte offset; this kernel has no static
// __shared__, and round-1 disassembly confirmed dynamic-LDS base == 0, so the
// asm byte offsets below match the smem[] pointer layout exactly.
// ---------------------------------------------------------------------------

#define DIN  64
#define UNI  128
#define CZ   192            // DIN + UNI
#define NN   36             // graph nodes
#define NP   48             // NN padded to multiple of 16
#define KCAT 768            // 4 * CZ
#define GS   50             // LDS row stride (floats) -- avoids bank conflicts
#define AS   50
#define PS   50
#define HS   50

// LDS byte-offset map (must mirror the smem[] pointer carving in the kernel).
#define LDS_AL 0u
#define LDS_G  (LDS_AL + 3u * NP * AS * 4u)     // 28800
#define LDS_P  (LDS_G + (unsigned)KCAT * GS * 4u)
#define LDS_H  (LDS_P + 256u * PS * 4u)
#define LDS_BI (LDS_H + (unsigned)UNI * HS * 4u)
#define LDS_BYTES (LDS_BI + 384u * 4u)          // 260736 B < 320KB WGP LDS

typedef float v2f __attribute__((ext_vector_type(2)));
typedef float v8f __attribute__((ext_vector_type(8)));

__device__ __forceinline__ v8f wmma_f32(v2f a, v2f b, v8f c) {
  // D(16x16 f32) = A(16x4 f32) * B(4x16 f32) + C
  return __builtin_amdgcn_wmma_f32_16x16x4_f32(
      /*neg_a=*/false, a, /*neg_b=*/false, b,
      /*c_mod=*/(short)0, c, /*reuse_a=*/false, /*reuse_b=*/false);
}

// Async copy: 16 bytes per lane, global -> LDS, tracked by ASYNCcnt.
__device__ __forceinline__ void async_g2l_b128(unsigned lds_off, const void* g) {
  asm volatile("global_load_async_to_lds_b128 %0, %1, off"
               :: "v"(lds_off), "v"((unsigned long long)(uintptr_t)g)
               : "memory");
}
// Async copy: 16 bytes per lane, LDS -> global, tracked by ASYNCcnt.
__device__ __forceinline__ void async_l2g_b128(void* g, unsigned lds_off) {
  asm volatile("global_store_async_from_lds_b128 %0, %1, off"
               :: "v"((unsigned long long)(uintptr_t)g), "v"(lds_off)
               : "memory");
}
__device__ __forceinline__ void wait_async0() {
  asm volatile("s_wait_asynccnt 0x0" ::: "memory");
}

// A-operand (16x4, f32): lane L -> row M = L&15, VGPR0 = K(kbase+2*(L>>4)),
// VGPR1 = K(+1). Contiguous pair -> single 8B LDS load.
__device__ __forceinline__ v2f load_a_lds(const float* M, int stride,
                                          int mbase, int kbase, int lane) {
  const float* p = M + (mbase + (lane & 15)) * stride + kbase + ((lane >> 4) << 1);
  v2f r = {p[0], p[1]};
  return r;
}

// B-operand (4x16, f32): lane L -> col N = L&15, VGPR0 = row kbase+2*(L>>4),
// VGPR1 = next row (mirrors A/C striping). Lowers to ds_load_2addr_b32.
__device__ __forceinline__ v2f load_b_lds(const float* M, int stride,
                                          int kbase, int nbase, int lane) {
  int k0 = kbase + ((lane >> 4) << 1);
  int n  = nbase + (lane & 15);
  v2f r = {M[k0 * stride + n], M[(k0 + 1) * stride + n]};
  return r;
}

// Diffusion: G[CZ + k*CZ + m][n] = (Z * A_{k+1})[m][n], Z = G rows 0..191.
// 3 powers * 12 Mtiles * 3 Ntiles = 108 WMMA C-tiles over 8 waves.
__device__ __forceinline__ void diffusion_wmma(float* G, const float* Al,
                                               int wave, int lane) {
  for (int t = wave; t < 108; t += 8) {
    int k   = t / 36;
    int rem = t - k * 36;
    int mt  = rem / 3;
    int nt  = rem - mt * 3;
    const float* Ak = Al + k * NP * AS;
    v8f acc = {0.f, 0.f, 0.f, 0.f, 0.f, 0.f, 0.f, 0.f};
    for (int ks = 0; ks < NP / 4; ++ks) {
      v2f a = load_a_lds(G, GS, mt * 16, ks * 4, lane);
      v2f b = load_b_lds(Ak, AS, ks * 4, nt * 16, lane);
      acc = wmma_f32(a, b, acc);
    }
    // C/D layout: VGPR i -> row i (lanes 0-15) / row 8+i (lanes 16-31)
    int n  = nt * 16 + (lane & 15);
    int mo = CZ + k * CZ + mt * 16 + ((lane >> 4) << 3);
    for (int i = 0; i < 8; ++i) G[(mo + i) * GS + n] = acc[i];
  }
}

// ---------------------------------------------------------------------------
// Kernel 1: A1 = adj^T, A2 = A1*A1, A3 = A2*A1, zero-padded [3][48][50] in ws.
// Tiny (36^3 MACs); runs once, consumed by every block of kernel 2.
// ---------------------------------------------------------------------------
__global__ void adj_powers_kernel(const float* __restrict__ adj,
                                  float* __restrict__ Apow) {
  __shared__ float A1s[NP * AS];
  __shared__ float A2s[NP * AS];
  int tid = threadIdx.x;
  for (int i = tid; i < NP * AS; i += blockDim.x) {
    int m = i / AS, n = i % AS;
    A1s[i] = (m < NN && n < NN) ? adj[n * NN + m] : 0.f;  // transpose
  }
  __syncthreads();
  for (int i = tid; i < NP * AS; i += blockDim.x) {
    int m = i / AS, n = i % AS;
    float s = 0.f;
    if (m < NN && n < NN)
      for (int p = 0; p < NN; ++p) s += A1s[m * AS + p] * A1s[p * AS + n];
    A2s[i] = s;
  }
  __syncthreads();
  for (int i = tid; i < NP * AS; i += blockDim.x) {
    int m = i / AS, n = i % AS;
    Apow[0 * NP * AS + i] = A1s[i];
    Apow[1 * NP * AS + i] = A2s[i];
    float s = 0.f;
    if (m < NN && n < NN)
      for (int p = 0; p < NN; ++p) s += A2s[m * AS + p] * A1s[p * AS + n];
    Apow[2 * NP * AS + i] = s;
  }
}

// ---------------------------------------------------------------------------
// Kernel 2: fully fused GCGRU cell, one block (8 waves) per batch element.
// ---------------------------------------------------------------------------
__global__ void __launch_bounds__(256) gcgru_kernel(
    const float* __restrict__ x,  const float* __restrict__ h,
    const float* __restrict__ Wf, const float* __restrict__ bf,
    const float* __restrict__ Wu, const float* __restrict__ bu,
    const float* __restrict__ Wc, const float* __restrict__ bc,
    const float* __restrict__ Apow, float* __restrict__ out) {
  extern __shared__ float smem[];
  float* Al = smem;                  // [3][48][AS]   adjacency powers
  float* G  = Al + 3 * NP * AS;      // [768][GS]     Z rows 0..191, diffusion 192..767
  float* P  = G + KCAT * GS;         // [256][PS]     r (0..127), u (128..255)
  float* H  = P + 256 * PS;          // [128][HS]     saved h
  float* Bi = H + UNI * HS;          // [384]         b_f | b_u | b_c

  const int tid  = threadIdx.x;
  const int lane = tid & 31;
  const int wave = tid >> 5;
  const int b    = blockIdx.x;

  // ---- phase 0a: zero pad-sensitive regions (plain DS stores) ----
  for (int i = tid; i < CZ * GS;  i += 256) G[i] = 0.f;
  for (int i = tid; i < UNI * HS; i += 256) H[i] = 0.f;
  for (int i = tid; i < 384; i += 256)
    Bi[i] = (i < 128) ? bf[i] : (i < 256) ? bu[i - 128] : bc[i - 256];
  __syncthreads();

  // ---- phase 0b: async global->LDS staging (ASYNCcnt path) ----
  // Adjacency powers: 28800 B = 1800 x 16B chunks into Al.
  {
    const char* ap = (const char*)Apow;
    for (int i = tid; i < 3 * NP * AS / 4; i += 256)
      async_g2l_b128(LDS_AL + (unsigned)i * 16u, ap + (size_t)i * 16);
  }
  // x rows: 36 floats = 144 B = 9 x 16B chunks; land at [c][0..35] of G.
  {
    const char* xb = (const char*)x + (size_t)b * DIN * NN * 4;
    for (int i = tid; i < DIN * 9; i += 256) {
      int c = i / 9, j = i % 9;
      async_g2l_b128(LDS_G + (unsigned)(c * GS * 4 + j * 16),
                     xb + (size_t)c * 144 + (size_t)j * 16);
    }
  }
  // h rows: into G rows 64..191 and (second async read, L2-hit) into H.
  {
    const char* hb = (const char*)h + (size_t)b * UNI * NN * 4;
    for (int i = tid; i < UNI * 9; i += 256) {
      int c = i / 9, j = i % 9;
      const char* src = hb + (size_t)c * 144 + (size_t)j * 16;
      async_g2l_b128(LDS_G + (unsigned)((DIN + c) * GS * 4 + j * 16), src);
      async_g2l_b128(LDS_H + (unsigned)(c * HS * 4 + j * 16), src);
    }
  }
  wait_async0();
  __syncthreads();

  // ---- phase 1: diffusion for the gate input (G = [z, zA, zA^2, zA^3]) ----
  diffusion_wmma(G, Al, wave, lane);
  __syncthreads();

  // ---- phase 2: gates pre = [W_f; W_u] * G  (M=256, K=768, N=48) ----
  // Each wave owns 2 M-tiles; A operand (weights) loaded once per k-step and
  // reused across all 3 N-tiles -> W read exactly once per block (L2 resident).
  for (int mi = 0; mi < 2; ++mi) {
    int mt  = wave * 2 + mi;
    int o16 = mt * 16;
    const float* W = (o16 < UNI) ? (Wf + (size_t)o16 * KCAT)
                                 : (Wu + (size_t)(o16 - UNI) * KCAT);
    const float* wrow = W + (size_t)(lane & 15) * KCAT + ((lane >> 4) << 1);
    __builtin_prefetch(wrow, 0, 1);
    v8f a0 = {0.f,0.f,0.f,0.f,0.f,0.f,0.f,0.f};
    v8f a1 = a0, a2 = a0;
    for (int ks = 0; ks < KCAT / 4; ++ks) {
      v2f av = {wrow[ks * 4], wrow[ks * 4 + 1]};
      v2f b0 = load_b_lds(G, GS, ks * 4,  0, lane);
      v2f b1 = load_b_lds(G, GS, ks * 4, 16, lane);
      v2f b2 = load_b_lds(G, GS, ks * 4, 32, lane);
      a0 = wmma_f32(av, b0, a0);
      a1 = wmma_f32(av, b1, a1);
      a2 = wmma_f32(av, b2, a2);
    }
    int n  = lane & 15;
    int mo = (lane >> 4) << 3;
    for (int i = 0; i < 8; ++i) {
      int o = o16 + mo + i;
      float bias = Bi[o];
      P[o * PS +  0 + n] = 1.f / (1.f + expf(-(a0[i] + bias)));
      P[o * PS + 16 + n] = 1.f / (1.f + expf(-(a1[i] + bias)));
      P[o * PS + 32 + n] = 1.f / (1.f + expf(-(a2[i] + bias)));
    }
  }
  __syncthreads();

  // ---- phase 3: candidate input Zc = [x, r*h] (overwrite h rows of Z) ----
  for (int i = tid; i < UNI * GS; i += 256) {
    int c = i / GS, n = i % GS;
    G[(DIN + c) * GS + n] = P[c * PS + n] * H[c * HS + n];  // pad cols: h==0
  }
  __syncthreads();

  // ---- phase 4: diffusion for the candidate input ----
  diffusion_wmma(G, Al, wave, lane);
  __syncthreads();

  // ---- phase 5: c = tanh(W_c * Gc + b_c); out = u*h + (1-u)*c ----
  // Result is staged packed [128][36] into the (now dead) r-rows of P, then
  // streamed out with coalesced 16B/lane async LDS->global stores.
  float* Pst = P;  // rows 0..127 of P: 4608 floats, disjoint from u rows
  {
    int o16 = wave * 16;  // 8 waves cover M=128
    const float* wrow = Wc + (size_t)o16 * KCAT +
                        (size_t)(lane & 15) * KCAT + ((lane >> 4) << 1);
    __builtin_prefetch(wrow, 0, 1);
    v8f a0 = {0.f,0.f,0.f,0.f,0.f,0.f,0.f,0.f};
    v8f a1 = a0, a2 = a0;
    for (int ks = 0; ks < KCAT / 4; ++ks) {
      v2f av = {wrow[ks * 4], wrow[ks * 4 + 1]};
      v2f b0 = load_b_lds(G, GS, ks * 4,  0, lane);
      v2f b1 = load_b_lds(G, GS, ks * 4, 16, lane);
      v2f b2 = load_b_lds(G, GS, ks * 4, 32, lane);
      a0 = wmma_f32(av, b0, a0);
      a1 = wmma_f32(av, b1, a1);
      a2 = wmma_f32(av, b2, a2);
    }
    int n  = lane & 15;
    int mo = (lane >> 4) << 3;
    for (int nt = 0; nt < 3; ++nt) {
      v8f acc = (nt == 0) ? a0 : (nt == 1) ? a1 : a2;
      int ng = nt * 16 + n;
      if (ng < NN) {
        for (int i = 0; i < 8; ++i) {
          int o   = o16 + mo + i;
          float c = tanhf(acc[i] + Bi[256 + o]);
          float u = P[(UNI + o) * PS + ng];
          float hv = H[o * HS + ng];
          Pst[o * NN + ng] = u * hv + (1.f - u) * c;
        }
      }
    }
  }
  __syncthreads();

  // ---- phase 6: coalesced async store of the contiguous 18432B out tile ----
  {
    char* ob = (char*)out + (size_t)b * UNI * NN * 4;
    for (int i = tid; i < UNI * NN / 4; i += 256)  // 1152 x 16B chunks
      async_l2g_b128(ob + (size_t)i * 16, LDS_P + (unsigned)i * 16u);
    wait_async0();
  }
}

// ---------------------------------------------------------------------------
extern "C" void kernel_launch(void* const* d_in, const int* in_sizes, int n_in,
                              void* d_out, int out_size, void* d_ws, size_t ws_size,
                              hipStream_t stream) {
  (void)n_in; (void)out_size; (void)ws_size;
  const float* x   = (const float*)d_in[0];
  const float* h   = (const float*)d_in[1];
  const float* adj = (const float*)d_in[2];
  const float* Wf  = (const float*)d_in[3];
  const float* bf  = (const float*)d_in[4];
  const float* Wu  = (const float*)d_in[5];
  const float* bu  = (const float*)d_in[6];
  const float* Wc  = (const float*)d_in[7];
  const float* bc  = (const float*)d_in[8];
  float* Apow = (float*)d_ws;               // [3][48][AS] adjacency powers
  float* out  = (float*)d_out;

  int B = in_sizes[0] / (DIN * NN);         // 4096

  adj_powers_kernel<<<1, 256, 0, stream>>>(adj, Apow);
  gcgru_kernel<<<B, 256, LDS_BYTES, stream>>>(x, h, Wf, bf, Wu, bu, Wc, bc,
                                              Apow, out);
}